// MagnusSemanticOperator_37048387895882
// MI455X (gfx1250) — compile-verified
//
#include <hip/hip_runtime.h>
#include <math.h>

typedef __attribute__((ext_vector_type(2))) float v2f;
typedef __attribute__((ext_vector_type(8))) float v8f;

#define B_ 32
#define N_ 1024
#define D_ 1024
#define O_ 1024
#define H_ 512
#define R_ 64
#define TOPK_ 4
#define PADX 4
#define LDW (64 + PADX)

// ---------------------------------------------------------------- K1: column mean of x[:,1:]
__global__ __launch_bounds__(256) void k_pm(const float* __restrict__ x, float* __restrict__ pm) {
  const int b = blockIdx.x;
  const int d = blockIdx.y * 256 + threadIdx.x;
  const float* xb = x + (size_t)b * N_ * D_;
  float acc = 0.f;
  for (int n = 1; n < N_; ++n) acc += xb[(size_t)n * D_ + d];
  pm[b * D_ + d] = acc * (1.0f / (float)(N_ - 1));
}

// ---------------------------------------------------------------- K2: proj = x @ basis^T (WMMA f32) + per-token ssq & dot(pm)
__global__ __launch_bounds__(256) void k_proj(const float* __restrict__ x,
                                              const float* __restrict__ basis,
                                              const float* __restrict__ pm,
                                              float* __restrict__ proj,
                                              float* __restrict__ ssq,
                                              float* __restrict__ dot) {
  __shared__ float xs[32 * LDW];
  __shared__ float bs[64 * LDW];
  const int b  = blockIdx.x;
  const int n0 = blockIdx.y * 32;
  const int t  = threadIdx.x;
  const int wave = t >> 5, lane = t & 31;
  const int ln = lane & 15, hi = lane >> 4;
  const int tw = wave >> 2;   // token sub-tile 0..1
  const int rw = wave & 3;    // r sub-tile 0..3
  const float* xb = x + ((size_t)b * N_ + n0) * D_;
  v8f acc = {};
  float ss = 0.f, dt = 0.f;
  const int tk8 = t >> 3, part = t & 7;

  for (int d0 = 0; d0 < D_; d0 += 64) {
    for (int i = 0; i < 8; ++i) {            // stage 32x64 x-chunk
      int l = t + i * 256;
      int tok = l >> 6, f = l & 63;
      xs[tok * LDW + f] = xb[(size_t)tok * D_ + d0 + f];
    }
    for (int i = 0; i < 16; ++i) {           // stage 64x64 basis chunk
      int l = t + i * 256;
      int r = l >> 6, f = l & 63;
      bs[r * LDW + f] = basis[(size_t)r * D_ + d0 + f];
    }
    __syncthreads();
    for (int j = 0; j < 8; ++j) {            // score partials
      float v = xs[tk8 * LDW + part * 8 + j];
      ss += v * v;
      dt += v * pm[b * D_ + d0 + part * 8 + j];
    }
    const float* arow = &xs[(tw * 16 + ln) * LDW];
    const float* brow = &bs[(rw * 16 + ln) * LDW];
    for (int kk = 0; kk < 64; kk += 4) {
      v2f a, bv;
      a.x  = arow[kk + 2 * hi + 0];
      a.y  = arow[kk + 2 * hi + 1];
      bv.x = brow[kk + 2 * hi + 0];
      bv.y = brow[kk + 2 * hi + 1];
      acc = __builtin_amdgcn_wmma_f32_16x16x4_f32(false, a, false, bv, (short)0, acc, false, false);
    }
    __syncthreads();
  }
  for (int off = 4; off >= 1; off >>= 1) {
    ss += __shfl_down(ss, off, 8);
    dt += __shfl_down(dt, off, 8);
  }
  if (part == 0) {
    ssq[b * N_ + n0 + tk8] = ss;
    dot[b * N_ + n0 + tk8] = dt;
  }
  const int col = rw * 16 + ln;
  for (int v = 0; v < 8; ++v)
    proj[((size_t)b * N_ + (n0 + tw * 16 + v + 8 * hi)) * R_ + col] = acc[v];
}

// ---------------------------------------------------------------- K3: top-4 tokens by (ssq - 2 dot), gather mean -> detail
__global__ __launch_bounds__(256) void k_topk(const float* __restrict__ x,
                                              const float* __restrict__ ssq,
                                              const float* __restrict__ dot,
                                              float* __restrict__ detail) {
  __shared__ float smax[256];
  __shared__ int   sidx[256];
  __shared__ int   chosen[TOPK_];
  const int b = blockIdx.x, t = threadIdx.x;
  for (int k = 0; k < TOPK_; ++k) {
    float bestv = -1e30f; int besti = 0x7fffffff;
    for (int j = t; j < N_ - 1; j += 256) {
      int n = j + 1;
      bool skip = false;
      for (int p = 0; p < k; ++p) if (chosen[p] == n) skip = true;
      float s = ssq[b * N_ + n] - 2.f * dot[b * N_ + n];
      if (!skip && (s > bestv || (s == bestv && n < besti))) { bestv = s; besti = n; }
    }
    smax[t] = bestv; sidx[t] = besti;
    __syncthreads();
    for (int off = 128; off >= 1; off >>= 1) {
      if (t < off) {
        if (smax[t + off] > smax[t] || (smax[t + off] == smax[t] && sidx[t + off] < sidx[t])) {
          smax[t] = smax[t + off]; sidx[t] = sidx[t + off];
        }
      }
      __syncthreads();
    }
    if (t == 0) chosen[k] = sidx[0];
    __syncthreads();
  }
  for (int d = t; d < D_; d += 256) {
    float s = 0.f;
    for (int p = 0; p < TOPK_; ++p) s += x[((size_t)b * N_ + chosen[p]) * D_ + d];
    detail[b * D_ + d] = s * 0.25f;
  }
}

// ---------------------------------------------------------------- K4: LN -> gelu(h@w1+b1) -> @w2+b2 -> fused=[context, local]
__global__ __launch_bounds__(256) void k_mlp(const float* __restrict__ detail,
                                             const float* __restrict__ gamma,
                                             const float* __restrict__ beta,
                                             const float* __restrict__ w1, const float* __restrict__ b1,
                                             const float* __restrict__ w2, const float* __restrict__ b2,
                                             const float* __restrict__ ctx, float* __restrict__ fused) {
  __shared__ float hn[D_];
  __shared__ float h1s[H_];
  __shared__ float red[256];
  const int b = blockIdx.x, t = threadIdx.x;
  float s = 0.f;
  for (int d = t; d < D_; d += 256) s += detail[b * D_ + d];
  red[t] = s; __syncthreads();
  for (int off = 128; off >= 1; off >>= 1) { if (t < off) red[t] += red[t + off]; __syncthreads(); }
  float mu = red[0] * (1.0f / D_);
  __syncthreads();
  float s2 = 0.f;
  for (int d = t; d < D_; d += 256) { float v = detail[b * D_ + d] - mu; s2 += v * v; }
  red[t] = s2; __syncthreads();
  for (int off = 128; off >= 1; off >>= 1) { if (t < off) red[t] += red[t + off]; __syncthreads(); }
  float inv = rsqrtf(red[0] * (1.0f / D_) + 1e-5f);
  __syncthreads();
  for (int d = t; d < D_; d += 256)
    hn[d] = (detail[b * D_ + d] - mu) * inv * gamma[d] + beta[d];
  __syncthreads();
  for (int h = t; h < H_; h += 256) {
    float a = b1[h];
    for (int k = 0; k < D_; ++k) a += hn[k] * w1[k * H_ + h];
    h1s[h] = 0.5f * a * (1.0f + erff(a * 0.70710678118654752f));
  }
  __syncthreads();
  for (int h = t; h < H_; h += 256) {
    float a = b2[h];
    for (int k = 0; k < H_; ++k) a += h1s[k] * w2[k * H_ + h];
    fused[b * 2 * H_ + H_ + h] = a;
    fused[b * 2 * H_ + h] = ctx[b * H_ + h];
  }
}

// ---------------------------------------------------------------- K5: ag = context@wg+bg ; al = local@wl+bl  (batched over all b)
__global__ __launch_bounds__(256) void k_skewgemm(const float* __restrict__ fused,
                                                  const float* __restrict__ wg, const float* __restrict__ bg,
                                                  const float* __restrict__ wl, const float* __restrict__ bl,
                                                  float* __restrict__ ag, float* __restrict__ al) {
  const int i = blockIdx.x * 256 + threadIdx.x;   // 0..4095
  const int which = blockIdx.y;                   // 0=global, 1=local
  const float* w = which ? wl : wg;
  const float* bias = which ? bl : bg;
  float acc[B_];
  for (int b = 0; b < B_; ++b) acc[b] = 0.f;
  for (int k = 0; k < H_; ++k) {
    float wv = w[(size_t)k * (R_ * R_) + i];
    for (int b = 0; b < B_; ++b)
      acc[b] += fused[b * 2 * H_ + which * H_ + k] * wv;
  }
  float bv = bias[i];
  float* outp = which ? al : ag;
  for (int b = 0; b < B_; ++b) outp[b * (R_ * R_) + i] = acc[b] + bv;
}

// ---------------------------------------------------------------- K6: mode_mask, gains, alpha/beta/gate
__global__ __launch_bounds__(256) void k_headvec(const float* __restrict__ fused,
    const float* __restrict__ wm, const float* __restrict__ bm,
    const float* __restrict__ wgain, const float* __restrict__ bgain,
    const float* __restrict__ wa, const float* __restrict__ ba,
    const float* __restrict__ wb, const float* __restrict__ bb,
    const float* __restrict__ wgate, const float* __restrict__ bgate,
    float* __restrict__ mm, float* __restrict__ gains, float* __restrict__ scal) {
  const int b = blockIdx.x, t = threadIdx.x;
  __shared__ float fs[2 * H_];
  for (int k = t; k < 2 * H_; k += 256) fs[k] = fused[b * 2 * H_ + k];
  __syncthreads();
  if (t < R_) {
    float a = bm[t];
    for (int k = 0; k < 2 * H_; ++k) a += fs[k] * wm[k * R_ + t];
    mm[b * R_ + t] = 1.f / (1.f + expf(-a));
  } else if (t < 2 * R_) {
    int r = t - R_;
    float a = bgain[r];
    for (int k = 0; k < 2 * H_; ++k) a += fs[k] * wgain[k * R_ + r];
    gains[b * R_ + r] = 1.0f + 0.1f * tanhf(a);
  } else if (t == 2 * R_) {
    float a = ba[0];
    for (int k = 0; k < 2 * H_; ++k) a += fs[k] * wa[k];
    scal[b * 4 + 0] = 0.5f * (1.f / (1.f + expf(-a)));
  } else if (t == 2 * R_ + 1) {
    float a = bb[0];
    for (int k = 0; k < 2 * H_; ++k) a += fs[k] * wb[k];
    scal[b * 4 + 1] = 0.25f * tanhf(a);
  } else if (t == 2 * R_ + 2) {
    float a = bgate[0];
    for (int k = 0; k < 2 * H_; ++k) a += fs[k] * wgate[k];
    scal[b * 4 + 2] = 1.f / (1.f + expf(-a));
  }
}

// ---------------------------------------------------------------- K7a: mask-scaled skews
__global__ __launch_bounds__(256) void k_mgml(const float* __restrict__ ag, const float* __restrict__ al,
                                              const float* __restrict__ mm,
                                              float* __restrict__ mg, float* __restrict__ ml) {
  const int b = blockIdx.x;
  const int idx = blockIdx.y * 256 + threadIdx.x;
  const int i = idx >> 6, j = idx & 63;
  const float m = mm[b * R_ + i] * mm[b * R_ + j] * 0.5f;
  const size_t base = (size_t)b * R_ * R_;
  mg[base + idx] = m * (ag[base + i * R_ + j] - ag[base + j * R_ + i]);
  ml[base + idx] = m * (al[base + i * R_ + j] - al[base + j * R_ + i]);
}

// ---------------------------------------------------------------- K7b: omega = a*0.5*(mg+ml) + b*[ml,mg]/12
__global__ __launch_bounds__(256) void k_omega(const float* __restrict__ mg, const float* __restrict__ ml,
                                               const float* __restrict__ scal, float* __restrict__ omega) {
  __shared__ float g[R_ * R_], l[R_ * R_];
  const int b = blockIdx.x, t = threadIdx.x;
  const size_t base = (size_t)b * R_ * R_;
  for (int i = t; i < R_ * R_; i += 256) { g[i] = mg[base + i]; l[i] = ml[base + i]; }
  __syncthreads();
  const float alpha = scal[b * 4 + 0], beta = scal[b * 4 + 1];
  for (int e = 0; e < 16; ++e) {
    int idx = t * 16 + e;
    int i = idx >> 6, j = idx & 63;
    float c = 0.f;
    for (int k = 0; k < R_; ++k) c += l[i * R_ + k] * g[k * R_ + j] - g[i * R_ + k] * l[k * R_ + j];
    omega[base + idx] = alpha * 0.5f * (g[idx] + l[idx]) + beta * (c * (1.0f / 12.0f));
  }
}

// ---------------------------------------------------------------- K8: expm via scaling(2^-6) + Taylor(10) + 6 squarings
__global__ __launch_bounds__(256) void k_expm(const float* __restrict__ omega, float* __restrict__ q) {
  __shared__ float T[R_ * R_], P[R_ * R_], Q[R_ * R_];
  const int b = blockIdx.x, t = threadIdx.x;
  const size_t base = (size_t)b * R_ * R_;
  for (int i = t; i < R_ * R_; i += 256) {
    float v = omega[base + i] * (1.0f / 64.0f);
    T[i] = v; P[i] = v;
    Q[i] = v + (((i >> 6) == (i & 63)) ? 1.0f : 0.0f);
  }
  __syncthreads();
  float regs[16];
  for (int kk = 2; kk <= 10; ++kk) {
    const float invk = 1.0f / (float)kk;
    for (int e = 0; e < 16; ++e) {
      int idx = t * 16 + e; int i = idx >> 6, j = idx & 63;
      float a = 0.f;
      for (int k = 0; k < R_; ++k) a += P[i * R_ + k] * T[k * R_ + j];
      regs[e] = a * invk;
    }
    __syncthreads();
    for (int e = 0; e < 16; ++e) { int idx = t * 16 + e; P[idx] = regs[e]; Q[idx] += regs[e]; }
    __syncthreads();
  }
  for (int sq = 0; sq < 6; ++sq) {
    for (int e = 0; e < 16; ++e) {
      int idx = t * 16 + e; int i = idx >> 6, j = idx & 63;
      float a = 0.f;
      for (int k = 0; k < R_; ++k) a += Q[i * R_ + k] * Q[k * R_ + j];
      regs[e] = a;
    }
    __syncthreads();
    for (int e = 0; e < 16; ++e) Q[t * 16 + e] = regs[e];
    __syncthreads();
  }
  for (int i = t; i < R_ * R_; i += 256) q[base + i] = Q[i];
}

// ---------------------------------------------------------------- K9: F[b] = gate_b * (Q*gains - I) @ output_basis
__global__ __launch_bounds__(256) void k_fop(const float* __restrict__ q, const float* __restrict__ gains,
                                             const float* __restrict__ scal, const float* __restrict__ ob,
                                             float* __restrict__ Fop) {
  __shared__ float M[R_ * R_];
  const int b = blockIdx.x, t = threadIdx.x;
  const size_t base = (size_t)b * R_ * R_;
  for (int i = t; i < R_ * R_; i += 256) {
    int r = i >> 6, c = i & 63;
    M[i] = q[base + i] * gains[b * R_ + c] - ((r == c) ? 1.0f : 0.0f);
  }
  __syncthreads();
  const int o = blockIdx.y * 256 + t;
  const float gate = scal[b * 4 + 2];
  float acc[R_];
  for (int r = 0; r < R_; ++r) acc[r] = 0.f;
  for (int k = 0; k < R_; ++k) {
    float v = ob[(size_t)k * O_ + o];
    for (int r = 0; r < R_; ++r) acc[r] += M[r * R_ + k] * v;
  }
  for (int r = 0; r < R_; ++r) Fop[((size_t)b * R_ + r) * O_ + o] = gate * acc[r];
}

// ---------------------------------------------------------------- K10: out[b] = proj[b] @ F[b]  (WMMA f32)
__global__ __launch_bounds__(256) void k_out(const float* __restrict__ proj,
                                             const float* __restrict__ Fop,
                                             float* __restrict__ out) {
  __shared__ float ps[32 * LDW];
  __shared__ float fs[64 * LDW];
  const int n0 = blockIdx.x * 32;
  const int o0 = blockIdx.y * 64;
  const int b  = blockIdx.z;
  const int t  = threadIdx.x;
  const int wave = t >> 5, lane = t & 31, ln = lane & 15, hi = lane >> 4;
  const int tw = wave >> 2, ow = wave & 3;
  for (int i = 0; i < 8; ++i) {
    int l = t + i * 256; int tok = l >> 6, r = l & 63;
    ps[tok * LDW + r] = proj[((size_t)b * N_ + n0 + tok) * R_ + r];
  }
  for (int i = 0; i < 16; ++i) {
    int l = t + i * 256; int r = l >> 6, oo = l & 63;
    fs[r * LDW + oo] = Fop[((size_t)b * R_ + r) * O_ + o0 + oo];
  }
  __syncthreads();
  v8f acc = {};
  const float* arow = &ps[(tw * 16 + ln) * LDW];
  for (int kk = 0; kk < 64; kk += 4) {
    v2f a, bv;
    a.x  = arow[kk + 2 * hi + 0];
    a.y  = arow[kk + 2 * hi + 1];
    bv.x = fs[(kk + 2 * hi + 0) * LDW + ow * 16 + ln];
    bv.y = fs[(kk + 2 * hi + 1) * LDW + ow * 16 + ln];
    acc = __builtin_amdgcn_wmma_f32_16x16x4_f32(false, a, false, bv, (short)0, acc, false, false);
  }
  for (int v = 0; v < 8; ++v)
    out[((size_t)b * N_ + n0 + tw * 16 + v + 8 * hi) * O_ + o0 + ow * 16 + ln] = acc[v];
}

// ================================================================ launch
extern "C" void kernel_launch(void* const* d_in, const int* in_sizes, int n_in,
                              void* d_out, int out_size, void* d_ws, size_t ws_size,
                              hipStream_t stream) {
  const float* x            = (const float*)d_in[0];
  const float* context      = (const float*)d_in[1];
  const float* input_basis  = (const float*)d_in[2];
  const float* output_basis = (const float*)d_in[3];
  const float* ln_gamma     = (const float*)d_in[4];
  const float* ln_beta      = (const float*)d_in[5];
  const float* w1 = (const float*)d_in[6];   const float* b1 = (const float*)d_in[7];
  const float* w2 = (const float*)d_in[8];   const float* b2 = (const float*)d_in[9];
  const float* wg = (const float*)d_in[10];  const float* bg = (const float*)d_in[11];
  const float* wl = (const float*)d_in[12];  const float* bl = (const float*)d_in[13];
  const float* wm = (const float*)d_in[14];  const float* bm = (const float*)d_in[15];
  const float* wgain = (const float*)d_in[16]; const float* bgain = (const float*)d_in[17];
  const float* wgate = (const float*)d_in[18]; const float* bgate = (const float*)d_in[19];
  const float* wa = (const float*)d_in[20];  const float* ba = (const float*)d_in[21];
  const float* wb = (const float*)d_in[22];  const float* bb = (const float*)d_in[23];
  float* out = (float*)d_out;
  float* ws  = (float*)d_ws;

  size_t off = 0;
  float* pm     = ws + off; off += (size_t)B_ * D_;
  float* ssq    = ws + off; off += (size_t)B_ * N_;
  float* dotv   = ws + off; off += (size_t)B_ * N_;
  float* detail = ws + off; off += (size_t)B_ * D_;
  float* fused  = ws + off; off += (size_t)B_ * 2 * H_;
  float* ag     = ws + off; off += (size_t)B_ * R_ * R_;
  float* al     = ws + off; off += (size_t)B_ * R_ * R_;
  float* mg     = ws + off; off += (size_t)B_ * R_ * R_;
  float* ml     = ws + off; off += (size_t)B_ * R_ * R_;
  float* mmv    = ws + off; off += (size_t)B_ * R_;
  float* gains  = ws + off; off += (size_t)B_ * R_;
  float* scal   = ws + off; off += (size_t)B_ * 4;
  float* omega  = ws + off; off += (size_t)B_ * R_ * R_;
  float* qm     = ws + off; off += (size_t)B_ * R_ * R_;
  float* Fop    = ws + off; off += (size_t)B_ * R_ * O_;
  float* proj   = ws + off; off += (size_t)B_ * N_ * R_;

  k_pm     <<<dim3(B_, D_ / 256), 256, 0, stream>>>(x, pm);
  k_proj   <<<dim3(B_, N_ / 32), 256, 0, stream>>>(x, input_basis, pm, proj, ssq, dotv);
  k_topk   <<<B_, 256, 0, stream>>>(x, ssq, dotv, detail);
  k_mlp    <<<B_, 256, 0, stream>>>(detail, ln_gamma, ln_beta, w1, b1, w2, b2, context, fused);
  k_skewgemm<<<dim3((R_ * R_) / 256, 2), 256, 0, stream>>>(fused, wg, bg, wl, bl, ag, al);
  k_headvec<<<B_, 256, 0, stream>>>(fused, wm, bm, wgain, bgain, wa, ba, wb, bb, wgate, bgate,
                                    mmv, gains, scal);
  k_mgml   <<<dim3(B_, (R_ * R_) / 256), 256, 0, stream>>>(ag, al, mmv, mg, ml);
  k_omega  <<<B_, 256, 0, stream>>>(mg, ml, scal, omega);
  k_expm   <<<B_, 256, 0, stream>>>(omega, qm);
  k_fop    <<<dim3(B_, O_ / 256), 256, 0, stream>>>(qm, gains, scal, output_basis, Fop);
  k_out    <<<dim3(N_ / 32, O_ / 64, B_), 256, 0, stream>>>(proj, Fop, out);
}